// HybridGATGCN_3470333575783
// MI455X (gfx1250) — compile-verified
//
#include <hip/hip_runtime.h>
#include <math.h>

typedef float v2f __attribute__((ext_vector_type(2)));
typedef float v8f __attribute__((ext_vector_type(8)));

#define NNODES 50000
#define NEDGES 600000
#define DH 128
#define DOUT 64
#define NEG_SLOPE 0.2f
#define BN_EPS 1e-5f

__device__ __forceinline__ unsigned f2ord(float f) {
    unsigned u = __float_as_uint(f);
    return (u & 0x80000000u) ? ~u : (u | 0x80000000u);
}
__device__ __forceinline__ float ord2f(unsigned u) {
    return (u & 0x80000000u) ? __uint_as_float(u & 0x7fffffffu) : __uint_as_float(~u);
}
__device__ __forceinline__ float lrelu(float x) { return x > 0.f ? x : NEG_SLOPE * x; }

// ---------------------------------------------------------------------------
// FP32 WMMA GEMM: C[M,Nc] = A[M,K] @ B[K,Nc], K=128 fixed.
// One wave owns a 16-wide column tile (B frags preloaded into 64 VGPRs and
// reused across M tiles) and processes TWO 16-row M tiles at a time with
// independent, interleaved accumulator chains. Wave/tile indices are forced
// into SGPRs via readfirstlane so all control flow around v_wmma is scalar
// (EXEC provably stays all-ones, as the ISA requires).
// ---------------------------------------------------------------------------
template <int K>
__global__ void gemm_wmma_f32(const float* __restrict__ A, const float* __restrict__ B,
                              float* __restrict__ C, int M_tiles, int Nc) {
    const int lane = threadIdx.x & 31;
    // wave index is wave-uniform; make it compiler-visibly scalar
    const int wave = __builtin_amdgcn_readfirstlane(
        (int)((blockIdx.x * blockDim.x + threadIdx.x) >> 5));
    const int tiles_n = Nc >> 4;
    const int tn = wave % tiles_n;   // SGPR
    const int wm = wave / tiles_n;   // SGPR
    const int row = lane & 15;
    const int kh  = lane >> 4;       // 0 -> K pair {0,1}, 1 -> {2,3}

    // Preload B column tile frags (K/4 steps x v2f) — W is cache-hot & tiny.
    v2f bfrag[K / 4];
#pragma unroll
    for (int kk = 0; kk < K / 4; ++kk) {
        const float* bp = B + (size_t)(kk * 4 + kh * 2) * Nc + tn * 16 + row;
        bfrag[kk].x = bp[0];
        bfrag[kk].y = bp[Nc];
    }

    const int wm_stride = ((int)(gridDim.x * blockDim.x) >> 5) / tiles_n;
    for (int tm = 2 * wm; tm < M_tiles; tm += 2 * wm_stride) {
        const float* ap0 = A + (size_t)(tm * 16 + row) * K + kh * 2;
        if (tm + 1 < M_tiles) {             // scalar branch (tm in SGPR)
            const float* ap1 = ap0 + (size_t)16 * K;
            v8f acc0 = {}, acc1 = {};
#pragma unroll
            for (int kk = 0; kk < K / 4; ++kk) {
                v2f a0 = *(const v2f*)(ap0 + kk * 4);
                v2f a1 = *(const v2f*)(ap1 + kk * 4);
                acc0 = __builtin_amdgcn_wmma_f32_16x16x4_f32(
                    false, a0, false, bfrag[kk], (short)0, acc0, false, false);
                acc1 = __builtin_amdgcn_wmma_f32_16x16x4_f32(
                    false, a1, false, bfrag[kk], (short)0, acc1, false, false);
            }
            float* cp0 = C + (size_t)(tm * 16 + kh * 8) * Nc + tn * 16 + row;
            float* cp1 = cp0 + (size_t)16 * Nc;
#pragma unroll
            for (int r = 0; r < 8; ++r) {
                cp0[(size_t)r * Nc] = acc0[r];
                cp1[(size_t)r * Nc] = acc1[r];
            }
        } else {                            // odd trailing tile
            v8f acc = {};
#pragma unroll
            for (int kk = 0; kk < K / 4; ++kk) {
                v2f a = *(const v2f*)(ap0 + kk * 4);
                acc = __builtin_amdgcn_wmma_f32_16x16x4_f32(
                    false, a, false, bfrag[kk], (short)0, acc, false, false);
            }
            float* cp = C + (size_t)(tm * 16 + kh * 8) * Nc + tn * 16 + row;
#pragma unroll
            for (int r = 0; r < 8; ++r) cp[(size_t)r * Nc] = acc[r];
        }
    }
}

// ---------------------------------------------------------------------------
// Utility / graph kernels
// ---------------------------------------------------------------------------
__global__ void k_fill(float* __restrict__ p, float v, int n) {
    int i = blockIdx.x * blockDim.x + threadIdx.x;
    if (i < n) p[i] = v;
}

__global__ void k_deg(const int* __restrict__ dst, float* __restrict__ deg, int E) {
    int e = blockIdx.x * blockDim.x + threadIdx.x;
    if (e < E) atomicAdd(&deg[dst[e]], 1.0f);
}

__global__ void k_rsqrt_inplace(float* __restrict__ p, int n) {
    int i = blockIdx.x * blockDim.x + threadIdx.x;
    if (i < n) p[i] = rsqrtf(p[i]);
}

// GCN edge aggregation: out[dst] += h[src] * dinv[src]*dinv[dst]  (float4 chunks)
__global__ void k_gcn_agg(const float* __restrict__ h, const int* __restrict__ src,
                          const int* __restrict__ dst, const float* __restrict__ dinv,
                          float* __restrict__ out, int E, int C) {
    int idx = blockIdx.x * blockDim.x + threadIdx.x;
    int chunks = C >> 2;
    if (idx >= E * chunks) return;
    int e = idx / chunks;
    int c4 = (idx - e * chunks) << 2;
    int s = src[e], d = dst[e];
    float nrm = dinv[s] * dinv[d];
    float4 hv = *(const float4*)(h + (size_t)s * C + c4);
    float* op = out + (size_t)d * C + c4;
    atomicAdd(op + 0, hv.x * nrm);
    atomicAdd(op + 1, hv.y * nrm);
    atomicAdd(op + 2, hv.z * nrm);
    atomicAdd(op + 3, hv.w * nrm);
}

// GCN finalize: self-loop term + bias + BN(eval) + ReLU
__global__ void k_gcn_fin(const float* __restrict__ agg, const float* __restrict__ h,
                          const float* __restrict__ dinv, const float* __restrict__ b,
                          const float* __restrict__ g, const float* __restrict__ be,
                          const float* __restrict__ mu, const float* __restrict__ var,
                          float* __restrict__ out, int nN, int C) {
    int idx = blockIdx.x * blockDim.x + threadIdx.x;
    if (idx >= nN * C) return;
    int n = idx / C, c = idx - n * C;
    float di = dinv[n];
    float v = agg[idx] + h[idx] * di * di + b[c];
    v = (v - mu[c]) * rsqrtf(var[c] + BN_EPS) * g[c] + be[c];
    out[idx] = fmaxf(v, 0.f);
}

// GAT per-node attention logits: al_s[n,h] = <hg[n,h,:], a_src[h,:]>
__global__ void k_gat_logits(const float* __restrict__ hg, const float* __restrict__ a_src,
                             const float* __restrict__ a_dst, float* __restrict__ als,
                             float* __restrict__ ald, int nN, int C) {
    int idx = blockIdx.x * blockDim.x + threadIdx.x;  // n*2 + head
    if (idx >= nN * 2) return;
    int n = idx >> 1, hh = idx & 1;
    const float* hp = hg + (size_t)n * (2 * C) + hh * C;
    const float* as = a_src + hh * C;
    const float* ad = a_dst + hh * C;
    float s = 0.f, d = 0.f;
    for (int c4 = 0; c4 < C; c4 += 4) {
        float4 x  = *(const float4*)(hp + c4);
        float4 ws = *(const float4*)(as + c4);
        float4 wd = *(const float4*)(ad + c4);
        s += x.x * ws.x + x.y * ws.y + x.z * ws.z + x.w * ws.w;
        d += x.x * wd.x + x.y * wd.y + x.z * wd.z + x.w * wd.w;
    }
    als[idx] = s;
    ald[idx] = d;
}

// Self-loop logit seeds the segment max (deg >= 1 with self-loops)
__global__ void k_gat_self_m(const float* __restrict__ als, const float* __restrict__ ald,
                             float* __restrict__ e_self, unsigned* __restrict__ m_ord, int n2) {
    int i = blockIdx.x * blockDim.x + threadIdx.x;
    if (i >= n2) return;
    float v = lrelu(als[i] + ald[i]);
    e_self[i] = v;
    m_ord[i] = f2ord(v);
}

__global__ void k_gat_edge_max(const float* __restrict__ als, const float* __restrict__ ald,
                               const int* __restrict__ src, const int* __restrict__ dst,
                               float* __restrict__ e_edge, unsigned* __restrict__ m_ord, int E) {
    int idx = blockIdx.x * blockDim.x + threadIdx.x;  // e*2 + head
    if (idx >= E * 2) return;
    int e = idx >> 1, hh = idx & 1;
    int s = src[e], d = dst[e];
    float v = lrelu(als[s * 2 + hh] + ald[d * 2 + hh]);
    e_edge[idx] = v;
    atomicMax(&m_ord[d * 2 + hh], f2ord(v));
}

__global__ void k_gat_self_den(const float* __restrict__ e_self, const unsigned* __restrict__ m_ord,
                               float* __restrict__ den, int n2) {
    int i = blockIdx.x * blockDim.x + threadIdx.x;
    if (i < n2) den[i] = expf(e_self[i] - ord2f(m_ord[i]));
}

__global__ void k_gat_edge_den(const float* __restrict__ e_edge, const unsigned* __restrict__ m_ord,
                               const int* __restrict__ dst, float* __restrict__ den, int E) {
    int idx = blockIdx.x * blockDim.x + threadIdx.x;
    if (idx >= E * 2) return;
    int e = idx >> 1, hh = idx & 1;
    int d = dst[e];
    atomicAdd(&den[d * 2 + hh], expf(e_edge[idx] - ord2f(m_ord[d * 2 + hh])));
}

// Weighted scatter: out[dst,c] += 0.5*(alpha0*hg[src,0,c] + alpha1*hg[src,1,c])
__global__ void k_gat_edge_agg(const float* __restrict__ hg, const float* __restrict__ e_edge,
                               const unsigned* __restrict__ m_ord, const float* __restrict__ den,
                               const int* __restrict__ src, const int* __restrict__ dst,
                               float* __restrict__ out, int E, int C) {
    int idx = blockIdx.x * blockDim.x + threadIdx.x;
    int chunks = C >> 2;
    if (idx >= E * chunks) return;
    int e = idx / chunks;
    int c4 = (idx - e * chunks) << 2;
    int s = src[e], d = dst[e];
    float a0 = 0.5f * expf(e_edge[e * 2 + 0] - ord2f(m_ord[d * 2 + 0])) / den[d * 2 + 0];
    float a1 = 0.5f * expf(e_edge[e * 2 + 1] - ord2f(m_ord[d * 2 + 1])) / den[d * 2 + 1];
    const float* h0 = hg + (size_t)s * (2 * C);
    float4 x0 = *(const float4*)(h0 + c4);
    float4 x1 = *(const float4*)(h0 + C + c4);
    float* op = out + (size_t)d * C + c4;
    atomicAdd(op + 0, a0 * x0.x + a1 * x1.x);
    atomicAdd(op + 1, a0 * x0.y + a1 * x1.y);
    atomicAdd(op + 2, a0 * x0.z + a1 * x1.z);
    atomicAdd(op + 3, a0 * x0.w + a1 * x1.w);
}

// GAT finalize: self-loop contribution + bias (+ optional ReLU)
__global__ void k_gat_fin(const float* __restrict__ agg, const float* __restrict__ hg,
                          const float* __restrict__ e_self, const unsigned* __restrict__ m_ord,
                          const float* __restrict__ den, const float* __restrict__ b,
                          float* __restrict__ out, int nN, int C, int do_relu) {
    int idx = blockIdx.x * blockDim.x + threadIdx.x;
    if (idx >= nN * C) return;
    int n = idx / C, c = idx - n * C;
    float a0 = 0.5f * expf(e_self[n * 2 + 0] - ord2f(m_ord[n * 2 + 0])) / den[n * 2 + 0];
    float a1 = 0.5f * expf(e_self[n * 2 + 1] - ord2f(m_ord[n * 2 + 1])) / den[n * 2 + 1];
    const float* h0 = hg + (size_t)n * (2 * C);
    float v = agg[idx] + a0 * h0[c] + a1 * h0[C + c] + b[c];
    out[idx] = do_relu ? fmaxf(v, 0.f) : v;
}

__global__ void k_logsoftmax64(const float* __restrict__ in, float* __restrict__ out, int nN) {
    int n = blockIdx.x * blockDim.x + threadIdx.x;
    if (n >= nN) return;
    const float* p = in + (size_t)n * 64;
    float4 v[16];
#pragma unroll
    for (int i = 0; i < 16; ++i) v[i] = *(const float4*)(p + i * 4);
    float m = -INFINITY;
#pragma unroll
    for (int i = 0; i < 16; ++i)
        m = fmaxf(m, fmaxf(fmaxf(v[i].x, v[i].y), fmaxf(v[i].z, v[i].w)));
    float s = 0.f;
#pragma unroll
    for (int i = 0; i < 16; ++i)
        s += expf(v[i].x - m) + expf(v[i].y - m) + expf(v[i].z - m) + expf(v[i].w - m);
    float ls = m + logf(s);
    float* o = out + (size_t)n * 64;
#pragma unroll
    for (int i = 0; i < 16; ++i) {
        float4 r = {v[i].x - ls, v[i].y - ls, v[i].z - ls, v[i].w - ls};
        *(float4*)(o + i * 4) = r;
    }
}

// ---------------------------------------------------------------------------
// Launcher
// ---------------------------------------------------------------------------
static inline int cdiv(long a, long b) { return (int)((a + b - 1) / b); }

extern "C" void kernel_launch(void* const* d_in, const int* in_sizes, int n_in,
                              void* d_out, int out_size, void* d_ws, size_t ws_size,
                              hipStream_t stream) {
    (void)in_sizes; (void)n_in; (void)out_size; (void)ws_size;
    const int N = NNODES, E = NEDGES;

    const float* x        = (const float*)d_in[0];
    const int*   ei       = (const int*)d_in[1];
    const int*   src      = ei;
    const int*   dst      = ei + E;
    const float* W_gcn1   = (const float*)d_in[2];
    const float* b_gcn1   = (const float*)d_in[3];
    const float* bn1_g    = (const float*)d_in[4];
    const float* bn1_b    = (const float*)d_in[5];
    const float* bn1_m    = (const float*)d_in[6];
    const float* bn1_v    = (const float*)d_in[7];
    const float* W_gat1   = (const float*)d_in[8];
    const float* a_src1   = (const float*)d_in[9];
    const float* a_dst1   = (const float*)d_in[10];
    const float* b_gat1   = (const float*)d_in[11];
    const float* W_gcn2   = (const float*)d_in[12];
    const float* b_gcn2   = (const float*)d_in[13];
    const float* bn2_g    = (const float*)d_in[14];
    const float* bn2_b    = (const float*)d_in[15];
    const float* bn2_m    = (const float*)d_in[16];
    const float* bn2_v    = (const float*)d_in[17];
    const float* W_gat2   = (const float*)d_in[18];
    const float* a_src2   = (const float*)d_in[19];
    const float* a_dst2   = (const float*)d_in[20];
    const float* b_gat2   = (const float*)d_in[21];

    // ---- workspace carve ----
    float* ws = (float*)d_ws;
    size_t off = 0;
    auto carve = [&](size_t nfloats) {
        float* p = ws + off;
        off += (nfloats + 63) & ~(size_t)63;
        return p;
    };
    float*    dinv    = carve(N);
    float*    buf_t   = carve((size_t)N * 256);   // GEMM output (h or hg)
    float*    buf_agg = carve((size_t)N * 128);   // scatter accumulator
    float*    buf_h   = carve((size_t)N * 128);   // stage features
    float*    als     = carve((size_t)N * 2);
    float*    ald     = carve((size_t)N * 2);
    unsigned* m_ord   = (unsigned*)carve((size_t)N * 2);
    float*    den     = carve((size_t)N * 2);
    float*    e_self  = carve((size_t)N * 2);
    float*    e_edge  = carve((size_t)E * 2);

    const int T = 256;
    const int M_tiles = N / 16;                    // 3125
    const dim3 blk(T);
    // GEMM: tiles_n * 1024 waves -> tiles_n * 128 blocks of 8 waves
    const dim3 g_gemm128(8 * 128), g_gemm256(16 * 128);

    // ---- degree / symmetric norm ----
    k_fill<<<cdiv(N, T), blk, 0, stream>>>(dinv, 1.0f, N);   // self-loop
    k_deg<<<cdiv(E, T), blk, 0, stream>>>(dst, dinv, E);
    k_rsqrt_inplace<<<cdiv(N, T), blk, 0, stream>>>(dinv, N);

    // ---- GCN1 ----
    gemm_wmma_f32<128><<<g_gemm128, blk, 0, stream>>>(x, W_gcn1, buf_t, M_tiles, 128);
    k_fill<<<cdiv((long)N * 128, T), blk, 0, stream>>>(buf_agg, 0.f, N * 128);
    k_gcn_agg<<<cdiv((long)E * 32, T), blk, 0, stream>>>(buf_t, src, dst, dinv, buf_agg, E, 128);
    k_gcn_fin<<<cdiv((long)N * 128, T), blk, 0, stream>>>(buf_agg, buf_t, dinv, b_gcn1,
                                                          bn1_g, bn1_b, bn1_m, bn1_v,
                                                          buf_h, N, 128);

    // ---- GAT1 (C=128, heads=2) ----
    gemm_wmma_f32<128><<<g_gemm256, blk, 0, stream>>>(buf_h, W_gat1, buf_t, M_tiles, 256);
    k_gat_logits<<<cdiv((long)N * 2, T), blk, 0, stream>>>(buf_t, a_src1, a_dst1, als, ald, N, 128);
    k_gat_self_m<<<cdiv((long)N * 2, T), blk, 0, stream>>>(als, ald, e_self, m_ord, N * 2);
    k_gat_edge_max<<<cdiv((long)E * 2, T), blk, 0, stream>>>(als, ald, src, dst, e_edge, m_ord, E);
    k_gat_self_den<<<cdiv((long)N * 2, T), blk, 0, stream>>>(e_self, m_ord, den, N * 2);
    k_gat_edge_den<<<cdiv((long)E * 2, T), blk, 0, stream>>>(e_edge, m_ord, dst, den, E);
    k_fill<<<cdiv((long)N * 128, T), blk, 0, stream>>>(buf_agg, 0.f, N * 128);
    k_gat_edge_agg<<<cdiv((long)E * 32, T), blk, 0, stream>>>(buf_t, e_edge, m_ord, den,
                                                              src, dst, buf_agg, E, 128);
    k_gat_fin<<<cdiv((long)N * 128, T), blk, 0, stream>>>(buf_agg, buf_t, e_self, m_ord, den,
                                                          b_gat1, buf_h, N, 128, 1);

    // ---- GCN2 ----
    gemm_wmma_f32<128><<<g_gemm128, blk, 0, stream>>>(buf_h, W_gcn2, buf_t, M_tiles, 128);
    k_fill<<<cdiv((long)N * 128, T), blk, 0, stream>>>(buf_agg, 0.f, N * 128);
    k_gcn_agg<<<cdiv((long)E * 32, T), blk, 0, stream>>>(buf_t, src, dst, dinv, buf_agg, E, 128);
    k_gcn_fin<<<cdiv((long)N * 128, T), blk, 0, stream>>>(buf_agg, buf_t, dinv, b_gcn2,
                                                          bn2_g, bn2_b, bn2_m, bn2_v,
                                                          buf_h, N, 128);

    // ---- GAT2 (C=64, heads=2 -> Nc = 128) ----
    gemm_wmma_f32<128><<<g_gemm128, blk, 0, stream>>>(buf_h, W_gat2, buf_t, M_tiles, 128);
    k_gat_logits<<<cdiv((long)N * 2, T), blk, 0, stream>>>(buf_t, a_src2, a_dst2, als, ald, N, 64);
    k_gat_self_m<<<cdiv((long)N * 2, T), blk, 0, stream>>>(als, ald, e_self, m_ord, N * 2);
    k_gat_edge_max<<<cdiv((long)E * 2, T), blk, 0, stream>>>(als, ald, src, dst, e_edge, m_ord, E);
    k_gat_self_den<<<cdiv((long)N * 2, T), blk, 0, stream>>>(e_self, m_ord, den, N * 2);
    k_gat_edge_den<<<cdiv((long)E * 2, T), blk, 0, stream>>>(e_edge, m_ord, dst, den, E);
    k_fill<<<cdiv((long)N * 64, T), blk, 0, stream>>>(buf_agg, 0.f, N * 64);
    k_gat_edge_agg<<<cdiv((long)E * 16, T), blk, 0, stream>>>(buf_t, e_edge, m_ord, den,
                                                              src, dst, buf_agg, E, 64);
    k_gat_fin<<<cdiv((long)N * 64, T), blk, 0, stream>>>(buf_agg, buf_t, e_self, m_ord, den,
                                                         b_gat2, buf_h, N, 64, 0);

    // ---- log_softmax over 64 features ----
    k_logsoftmax64<<<cdiv(N, T), blk, 0, stream>>>(buf_h, (float*)d_out, N);
}